// VisionAttention_35656818491947
// MI455X (gfx1250) — compile-verified
//
#include <hip/hip_runtime.h>
#include <math.h>

// VisionAttention on gfx1250: fp32 WMMA (V_WMMA_F32_16X16X4_F32) everywhere.
// N=4096 tokens, D=1280, H=16 heads, HD=80, 4 segments of 1024.

typedef __attribute__((ext_vector_type(2))) float v2f;
typedef __attribute__((ext_vector_type(8))) float v8f;

#define N_TOK 4096
#define DIM   1280
#define NH    16
#define HD    80
#define QKV_DIM 3840

__device__ __forceinline__ v8f wmma_f32(v2f a, v2f b, v8f c) {
  // 8 args: (neg_a, A, neg_b, B, c_mod, C, reuse_a, reuse_b)
  return __builtin_amdgcn_wmma_f32_16x16x4_f32(false, a, false, b, (short)0, c,
                                               false, false);
}

// ---------------------------------------------------------------------------
// QKV GEMM: Y = X @ Wqkv^T + b, scatter-stored into qkv_ws[t][h][m][hd].
// One wave per 32(M) x 64(N) tile: 2 A-fragments reuse 4 B-fragments
// (8 WMMA per 6 b64 loads per k-step).
// ---------------------------------------------------------------------------
__global__ void va_qkv_gemm(const float* __restrict__ X,
                            const float* __restrict__ W,
                            const float* __restrict__ bias,
                            float* __restrict__ qkv_ws) {
  const int lane = threadIdx.x & 31;
  const int r = lane & 15;
  const int h = lane >> 4;
  const int m0 = blockIdx.x * 32;
  const int n0 = blockIdx.y * 64;

  v8f acc[2][4] = {};
  const float* arow0 = X + (size_t)(m0 + r) * DIM + 2 * h;
  const float* arow1 = arow0 + (size_t)16 * DIM;
  const float* brow  = W + (size_t)(n0 + r) * DIM + 2 * h;

  for (int k = 0; k < DIM; k += 4) {
    v2f a0 = *(const v2f*)(arow0 + k);
    v2f a1 = *(const v2f*)(arow1 + k);
#pragma unroll
    for (int t = 0; t < 4; ++t) {
      v2f b = *(const v2f*)(brow + (size_t)(t * 16) * DIM + k);
      acc[0][t] = wmma_f32(a0, b, acc[0][t]);
      acc[1][t] = wmma_f32(a1, b, acc[1][t]);
    }
  }

#pragma unroll
  for (int t = 0; t < 4; ++t) {
    const int c = n0 + t * 16 + r;          // column in [0, 3840)
    const int tsel = c / DIM;               // 0=q 1=k 2=v
    const int d = c - tsel * DIM;
    const int head = d / HD;
    const int hd = d - head * HD;
    const float bv = bias[c];
    float* dst = qkv_ws + ((size_t)tsel * NH + head) * (size_t)N_TOK * HD + hd;
#pragma unroll
    for (int mm = 0; mm < 2; ++mm) {
#pragma unroll
      for (int j = 0; j < 8; ++j) {
        dst[(size_t)(m0 + mm * 16 + j + 8 * h) * HD] = acc[mm][t][j] + bv;
      }
    }
  }
}

// ---------------------------------------------------------------------------
// RoPE applied in-place to q (t=0) and k (t=1). One thread per (t,h,m,d<40).
// ---------------------------------------------------------------------------
__global__ void va_rope(float* __restrict__ qkv_ws,
                        const float* __restrict__ rope) {
  int idx = blockIdx.x * blockDim.x + threadIdx.x;
  const int total = 2 * NH * N_TOK * (HD / 2);
  if (idx >= total) return;
  const int d = idx % (HD / 2);
  int tmp = idx / (HD / 2);
  const int m = tmp % N_TOK;
  tmp /= N_TOK;
  const int head = tmp % NH;
  const int tsel = tmp / NH;

  const float ang = rope[m * (HD / 2) + d];
  const float c = cosf(ang);
  const float s = sinf(ang);
  float* base = qkv_ws + ((size_t)tsel * NH + head) * (size_t)N_TOK * HD +
                (size_t)m * HD;
  const float x1 = base[d];
  const float x2 = base[d + HD / 2];
  base[d] = x1 * c - x2 * s;
  base[d + HD / 2] = x2 * c + x1 * s;
}

// ---------------------------------------------------------------------------
// Flash attention, one wave per (16-query block, head). Q fragments are
// hoisted into registers (16x80 = 20 v2f) across the whole key loop.
// ---------------------------------------------------------------------------
__global__ void va_attn(const float* __restrict__ qkv_ws,
                        const int* __restrict__ cu, int ncu,
                        float* __restrict__ attn_out) {
  __shared__ float pbuf[16 * 16];   // one wave per block
  const int lane = threadIdx.x & 31;
  const int r = lane & 15;
  const int h = lane >> 4;
  const int q0 = blockIdx.x * 16;
  const int head = blockIdx.y;

  int lo = 0, hi = N_TOK;
  for (int i = 0; i + 1 < ncu; ++i) {
    const int a = cu[i], b = cu[i + 1];
    if (a <= q0 && q0 < b) { lo = a; hi = b; }
  }

  const float* Q  = qkv_ws + (size_t)head * N_TOK * HD;
  const float* Kp = qkv_ws + ((size_t)NH + head) * (size_t)N_TOK * HD;
  const float* Vp = qkv_ws + ((size_t)2 * NH + head) * (size_t)N_TOK * HD;

  const float scale = 0.11180339887498949f;  // 1/sqrt(80)

  // Hoist the Q fragment: loop-invariant across all key chunks.
  v2f qf[HD / 4];
  {
    const float* qrow = Q + (size_t)(q0 + r) * HD + 2 * h;
#pragma unroll
    for (int d = 0; d < HD / 4; ++d) qf[d] = *(const v2f*)(qrow + 4 * d);
  }

  v8f o[5] = {};
  float mi[8], li[8];
#pragma unroll
  for (int j = 0; j < 8; ++j) { mi[j] = -INFINITY; li[j] = 0.0f; }

  for (int k0 = lo; k0 < hi; k0 += 16) {
    // ---- S = (Q @ K^T) for a 16x16 tile --------------------------------
    v8f s = {};
    const float* krow = Kp + (size_t)(k0 + r) * HD + 2 * h;
#pragma unroll
    for (int d = 0; d < HD / 4; ++d) {
      v2f b = *(const v2f*)(krow + 4 * d);
      s = wmma_f32(qf[d], b, s);
    }

    // ---- online softmax: lane half h owns rows {8h..8h+7} --------------
    float p[8], alpha[8];
#pragma unroll
    for (int j = 0; j < 8; ++j) {
      float v = s[j] * scale;
      float rm = v;
#pragma unroll
      for (int off = 1; off < 16; off <<= 1)
        rm = fmaxf(rm, __shfl_xor(rm, off, 32));
      const float mnew = fmaxf(mi[j], rm);
      alpha[j] = __expf(mi[j] - mnew);
      const float pv = __expf(v - mnew);
      float rs = pv;
#pragma unroll
      for (int off = 1; off < 16; off <<= 1)
        rs += __shfl_xor(rs, off, 32);
      li[j] = li[j] * alpha[j] + rs;
      mi[j] = mnew;
      p[j] = pv;
    }
#pragma unroll
    for (int t = 0; t < 5; ++t)
#pragma unroll
      for (int j = 0; j < 8; ++j) o[t][j] *= alpha[j];

    // ---- re-layout P: C-fragment -> LDS -> A-fragment ------------------
    __syncthreads();
#pragma unroll
    for (int j = 0; j < 8; ++j) pbuf[(j + 8 * h) * 16 + r] = p[j];
    __syncthreads();

    // ---- O += P @ V ----------------------------------------------------
#pragma unroll
    for (int t = 0; t < 5; ++t) {
      const float* vcol = Vp + (size_t)k0 * HD + t * 16 + r;
#pragma unroll
      for (int kk = 0; kk < 16; kk += 4) {
        v2f a = *(const v2f*)(&pbuf[r * 16 + kk + 2 * h]);
        v2f b;
        b.x = vcol[(size_t)(kk + 2 * h) * HD];
        b.y = vcol[(size_t)(kk + 2 * h + 1) * HD];
        o[t] = wmma_f32(a, b, o[t]);
      }
    }
  }

  // ---- normalize and store: attn_out[m][head*80 + c] -------------------
#pragma unroll
  for (int t = 0; t < 5; ++t) {
    const int c = head * HD + t * 16 + r;
#pragma unroll
    for (int j = 0; j < 8; ++j) {
      attn_out[(size_t)(q0 + j + 8 * h) * DIM + c] = o[t][j] / li[j];
    }
  }
}

// ---------------------------------------------------------------------------
// Output projection: C = A @ Wp^T + b, row-major C. One wave per 32x64 tile.
// ---------------------------------------------------------------------------
__global__ void va_proj_gemm(const float* __restrict__ A,
                             const float* __restrict__ W,
                             const float* __restrict__ bias,
                             float* __restrict__ C) {
  const int lane = threadIdx.x & 31;
  const int r = lane & 15;
  const int h = lane >> 4;
  const int m0 = blockIdx.x * 32;
  const int n0 = blockIdx.y * 64;

  v8f acc[2][4] = {};
  const float* arow0 = A + (size_t)(m0 + r) * DIM + 2 * h;
  const float* arow1 = arow0 + (size_t)16 * DIM;
  const float* brow  = W + (size_t)(n0 + r) * DIM + 2 * h;

  for (int k = 0; k < DIM; k += 4) {
    v2f a0 = *(const v2f*)(arow0 + k);
    v2f a1 = *(const v2f*)(arow1 + k);
#pragma unroll
    for (int t = 0; t < 4; ++t) {
      v2f b = *(const v2f*)(brow + (size_t)(t * 16) * DIM + k);
      acc[0][t] = wmma_f32(a0, b, acc[0][t]);
      acc[1][t] = wmma_f32(a1, b, acc[1][t]);
    }
  }

#pragma unroll
  for (int t = 0; t < 4; ++t) {
    const int c = n0 + t * 16 + r;
    const float bv = bias[c];
#pragma unroll
    for (int mm = 0; mm < 2; ++mm) {
#pragma unroll
      for (int j = 0; j < 8; ++j) {
        C[(size_t)(m0 + mm * 16 + j + 8 * h) * DIM + c] = acc[mm][t][j] + bv;
      }
    }
  }
}

extern "C" void kernel_launch(void* const* d_in, const int* in_sizes, int n_in,
                              void* d_out, int out_size, void* d_ws,
                              size_t ws_size, hipStream_t stream) {
  const float* hidden = (const float*)d_in[0];
  const float* qkv_w  = (const float*)d_in[1];
  const float* qkv_b  = (const float*)d_in[2];
  const float* proj_w = (const float*)d_in[3];
  const float* proj_b = (const float*)d_in[4];
  const float* rope   = (const float*)d_in[5];
  const int*   cu     = (const int*)d_in[6];
  const int    ncu    = in_sizes[6];

  float* qkv_ws  = (float*)d_ws;                                // 3*16*4096*80 f32
  float* attn_ws = qkv_ws + (size_t)3 * NH * N_TOK * HD;        // 4096*1280 f32

  const dim3 wave(32);

  va_qkv_gemm<<<dim3(N_TOK / 32, QKV_DIM / 64), wave, 0, stream>>>(
      hidden, qkv_w, qkv_b, qkv_ws);

  const int rope_total = 2 * NH * N_TOK * (HD / 2);
  va_rope<<<(rope_total + 255) / 256, 256, 0, stream>>>(qkv_ws, rope);

  va_attn<<<dim3(N_TOK / 16, NH), wave, 0, stream>>>(qkv_ws, cu, ncu, attn_ws);

  va_proj_gemm<<<dim3(N_TOK / 32, DIM / 64), wave, 0, stream>>>(
      attn_ws, proj_w, proj_b, (float*)d_out);
}